// Mel_to_wf_38062000177292
// MI455X (gfx1250) — compile-verified
//
#include <hip/hip_runtime.h>
#include <math.h>

typedef __attribute__((ext_vector_type(2))) float v2f;
typedef __attribute__((ext_vector_type(8))) float v8f;
typedef __attribute__((ext_vector_type(4))) unsigned int v4u;
typedef __attribute__((ext_vector_type(8))) unsigned int v8u;

#define BATCH  8
#define TFRM   1024          // frames
#define NFFT   1024
#define HOPSZ  256
#define NFREQ  513
#define FPAD   576           // NFREQ padded to /64 (GEMM N-tile)
#define CPAD   1088          // 2*NFREQ padded to /64 ([Re|Im] combined)
#define NMEL   80
#define OUTL   261888        // HOP*(T-1)
#define YLEN   262912        // NFFT + HOP*(T-1)
#define MOMC   0.49748743718592964f  // 0.99/(1+0.99)
#define TWO_PI 6.28318530717958647692f

// B matrices are stored row-pair interleaved so a WMMA B-fragment
// {B[k][n], B[k+1][n]} is one aligned 8-byte LDS load:
//   element (k, n) of a K x N matrix lives at (k>>1)*2*ldb + 2*n + (k&1)
#define BIDX(k, n, ldb) (((long long)((k) >> 1) * 2 * (ldb)) + ((n) << 1) + ((k) & 1))

// ---------------------------------------------------------------------------
// TDM: issue a 2-D tile DMA  global -> LDS  (CDNA5 tensor_load_to_lds).
// Descriptor packing per CDNA5 ISA.
// tdim0/tdim1: tensor extents (elements) for OOB zero-fill; stride: row stride
// (elements); tile_w x tile_h tile written packed row-major at lds_addr, with
// optional LDS padding (pad_interval code: 2^(c+1) DWORDs, pad_amount: c+1 DW).
// ---------------------------------------------------------------------------
__device__ __forceinline__ void tdm_load_2d(
    unsigned lds_addr, const void* gptr,
    unsigned tdim0, unsigned tdim1, unsigned stride,
    unsigned tile_w, unsigned tile_h,
    unsigned pad_en, unsigned pad_interval, unsigned pad_amount)
{
    unsigned long long ga = (unsigned long long)gptr;
    v4u g0;
    g0[0] = 1u;                                             // count=1 (valid user D#)
    g0[1] = lds_addr;                                       // lds_addr [63:32]
    g0[2] = (unsigned)ga;                                   // global_addr lo
    g0[3] = ((unsigned)(ga >> 32) & 0x01FFFFFFu)            // global_addr hi (57b)
          | 0x80000000u;                                    // type=2 ("image")
    v8u g1;
    g1[0] = (2u << 16)                                      // data_size = 4B
          | (pad_en << 20) | (pad_interval << 22) | (pad_amount << 25);
    g1[1] = (tdim0 & 0xFFFFu) << 16;                        // tensor_dim0 lo16 @ [63:48]
    g1[2] = (tdim0 >> 16) | ((tdim1 & 0xFFFFu) << 16);      // dim0 hi / dim1 lo
    g1[3] = (tdim1 >> 16) | (tile_w << 16);                 // dim1 hi / tile_dim0
    g1[4] = tile_h & 0xFFFFu;                               // tile_dim1 (tile_dim2=0)
    g1[5] = stride;                                         // tensor_dim0_stride lo32
    g1[6] = 0u;                                             // stride hi / dim1_stride lo
    g1[7] = 0u;
    // 2-operand form: VADDR2/VADDR3 = NULL (<=2D tensor)
    asm volatile("tensor_load_to_lds %0, %1" :: "s"(g0), "s"(g1) : "memory");
}

// ---------------------------------------------------------------------------
// fp32 WMMA GEMM: C[M,N] = (relu?)(A[M,K] @ B_int[K,N]); batched over
// gridDim.z on A and C (B shared, row-pair interleaved layout, see BIDX).
// Block = 4 waves; macro tile 64(M) x 64(N); wave w owns rows [16w,16w+16) x
// all 64 cols (4 accumulators -> 4x A-fragment reuse). K staged in chunks of
// 32 through double-buffered LDS filled by the Tensor Data Mover (wave 0
// issues, s_wait_tensorcnt(2) overlaps next-chunk DMA with current WMMAs).
// M%64==0, N%64==0, K%16==0 (K even), leading dims multiple of 4.
// ---------------------------------------------------------------------------
__global__ __launch_bounds__(128) void wmma_gemm_f32(
    const float* __restrict__ A, const float* __restrict__ B,
    float* __restrict__ C, int M, int N, int K,
    int lda, int ldb, int ldc, long long sA, long long sC, int relu)
{
    __shared__ __align__(16) float As[2][64][36];      // TDM-padded rows (32+4)
    __shared__ __align__(16) float Bs[2][16][64][2];   // [kpair][n][k&1]

    const int tid  = threadIdx.x;
    const int wave = tid >> 5;
    const int lane = tid & 31;
    const int mBlock = blockIdx.y * 64;
    const int n0 = blockIdx.x * 64;
    A += (long long)blockIdx.z * sA;
    C += (long long)blockIdx.z * sC;

    v8f acc[4];
    #pragma unroll
    for (int j = 0; j < 4; ++j) acc[j] = (v8f){0.f,0.f,0.f,0.f,0.f,0.f,0.f,0.f};

    const int klo  = (lane >> 4) << 1;   // 0 for lanes 0-15, 2 for 16-31
    const int kq0  = (lane >> 4);        // pair index offset (klo/2)
    const int r    = lane & 15;
    const int mrow = wave << 4;

    auto issue = [&](int kc, int buf) {
        unsigned rem = (unsigned)(K - kc);
        // A tile: 64 rows x 32 cols, LDS rows padded 32->36 floats by TDM
        tdm_load_2d((unsigned)(uintptr_t)&As[buf][0][0],
                    A + (long long)mBlock * lda + kc,
                    rem, 64u, (unsigned)lda, 32u, 64u,
                    1u, 4u, 3u);                 // pad every 32 DW by 4 DW
        // B tile (interleaved): 16 pair-rows x 128 floats, packed
        tdm_load_2d((unsigned)(uintptr_t)&Bs[buf][0][0][0],
                    B + (long long)kc * ldb + (n0 << 1),
                    128u, rem >> 1, (unsigned)(ldb << 1), 128u, 16u,
                    0u, 0u, 0u);
    };

    if (wave == 0) issue(0, 0);
    int buf = 0;
    for (int kc = 0; kc < K; kc += 32) {
        if (wave == 0) {
            if (kc + 32 < K) {
                issue(kc + 32, buf ^ 1);                  // prefetch next chunk
                __builtin_amdgcn_s_wait_tensorcnt(2);     // current chunk done
            } else {
                __builtin_amdgcn_s_wait_tensorcnt(0);
            }
        }
        __syncthreads();                                  // tile visible to all waves

        const int ks = (K - kc < 32) ? ((K - kc) >> 2) : 8;
        for (int kk = 0; kk < ks; ++kk) {
            const int kb = (kk << 2) + klo;
            const int kq = (kk << 1) + kq0;
            v2f a = *reinterpret_cast<const v2f*>(&As[buf][mrow + r][kb]);
            #pragma unroll
            for (int j = 0; j < 4; ++j) {
                v2f b = *reinterpret_cast<const v2f*>(&Bs[buf][kq][(j << 4) + r][0]);
                acc[j] = __builtin_amdgcn_wmma_f32_16x16x4_f32(
                             false, a, false, b, (short)0, acc[j], false, false);
            }
        }
        __syncthreads();                                  // done before TDM rewrites buf
        buf ^= 1;
    }

    const int crow = (lane >> 4) << 3;   // C: vgpr v -> M = crow+v, N = lane&15
    #pragma unroll
    for (int j = 0; j < 4; ++j)
        #pragma unroll
        for (int v = 0; v < 8; ++v) {
            float x = acc[j][v];
            if (relu) x = fmaxf(x, 0.f);
            C[(long long)(mBlock + mrow + crow + v) * ldc + n0 + (j << 4) + r] = x;
        }
}

// ------------------------ setup: mel filterbank & pinv ----------------------
__device__ __forceinline__ float mel_hz(float mel) {
    return 700.f * (expf(mel * (2.302585092994046f / 2595.f)) - 1.f);
}

__global__ void build_fb(float* __restrict__ fb) {   // fb[513][80]
    int idx = blockIdx.x * 256 + threadIdx.x;
    if (idx >= NFREQ * NMEL) return;
    int f = idx / NMEL, m = idx % NMEL;
    float mel_max = 2595.f * log10f(1.f + 8000.f / 700.f);
    float f0 = mel_hz(mel_max * (float)(m)     / 81.f);
    float f1 = mel_hz(mel_max * (float)(m + 1) / 81.f);
    float f2 = mel_hz(mel_max * (float)(m + 2) / 81.f);
    float freq = (float)f * (8000.f / 512.f);
    float down = (freq - f0) / (f1 - f0);
    float up   = (f2 - freq) / (f2 - f1);
    fb[idx] = fmaxf(0.f, fminf(down, up));
}

// single-block Gauss-Jordan inverse of G = fb^T fb (80x80, SPD)
__global__ __launch_bounds__(256) void invert_gram(
    const float* __restrict__ fb, float* __restrict__ invG)
{
    __shared__ float Aug[NMEL][161];
    __shared__ float fac[NMEL];
    int tid = threadIdx.x;
    for (int e = tid; e < NMEL * NMEL; e += 256) {
        int i = e / NMEL, j = e % NMEL;
        float s = 0.f;
        for (int f = 0; f < NFREQ; ++f)
            s += fb[f * NMEL + i] * fb[f * NMEL + j];
        Aug[i][j] = s;
        Aug[i][NMEL + j] = (i == j) ? 1.f : 0.f;
    }
    __syncthreads();
    for (int k = 0; k < NMEL; ++k) {
        float inv = 1.f / Aug[k][k];
        __syncthreads();
        for (int c = tid; c < 160; c += 256) Aug[k][c] *= inv;
        __syncthreads();
        for (int r2 = tid; r2 < NMEL; r2 += 256) fac[r2] = Aug[r2][k];
        __syncthreads();
        for (int e = tid; e < NMEL * 160; e += 256) {
            int r = e / 160, c = e % 160;
            if (r != k) Aug[r][c] -= fac[r] * Aug[k][c];
        }
        __syncthreads();
    }
    for (int e = tid; e < NMEL * NMEL; e += 256)
        invG[e] = Aug[e / NMEL][NMEL + e % NMEL];
}

// melB(k=m, n=f) = sum_j invG[m][j] * fb[f][j]  (= pinv(fb.T)^T), interleaved
__global__ void make_melB(const float* __restrict__ fb,
                          const float* __restrict__ invG,
                          float* __restrict__ melB) {
    int idx = blockIdx.x * 256 + threadIdx.x;
    if (idx >= NMEL * FPAD) return;
    int m = idx / FPAD, f = idx % FPAD;
    float v = 0.f;
    if (f < NFREQ)
        for (int k = 0; k < NMEL; ++k)
            v += invG[m * NMEL + k] * fb[f * NMEL + k];
    melB[BIDX(m, f, FPAD)] = v;
}

// ------------------------ setup: DFT matrices, window -----------------------
// Forward: rebuilt[t][c] = sum_n frames[t][n] * Cf(n, c)   (interleaved B)
__global__ void build_cf(float* __restrict__ Cf) {   // K=1024 x N=CPAD
    int idx = blockIdx.x * 256 + threadIdx.x;
    if (idx >= NFFT * CPAD) return;
    int n = idx / CPAD, c = idx % CPAD;
    float v = 0.f;
    if (c < 2 * NFREQ) {
        int f = (c < NFREQ) ? c : c - NFREQ;
        float ang = TWO_PI * (float)((f * n) & 1023) / 1024.f;
        float s, co; sincosf(ang, &s, &co);
        v = (c < NFREQ) ? co : -s;
    }
    Cf[BIDX(n, c, CPAD)] = v;
}

// Inverse: time[t][n] = sum_c SA[t][c] * Ci(c, n)   (irfft, Hermitian)
__global__ void build_ci(float* __restrict__ Ci) {   // K=CPAD x N=1024
    int idx = blockIdx.x * 256 + threadIdx.x;
    if (idx >= CPAD * NFFT) return;
    int r = idx / NFFT, n = idx % NFFT;
    float v = 0.f;
    if (r < NFREQ) {
        int f = r;
        float cfac = (f == 0 || f == NFREQ - 1) ? 1.f : 2.f;
        float ang = TWO_PI * (float)((f * n) & 1023) / 1024.f;
        v = cfac * cosf(ang) * (1.f / 1024.f);
    } else if (r < 2 * NFREQ) {
        int f = r - NFREQ;
        float ang = TWO_PI * (float)((f * n) & 1023) / 1024.f;
        v = -2.f * sinf(ang) * (1.f / 1024.f);
    }
    Ci[BIDX(r, n, NFFT)] = v;
}

__global__ void build_window(float* __restrict__ W) {
    int n = blockIdx.x * 256 + threadIdx.x;
    if (n >= NFFT) return;
    W[n] = 0.5f - 0.5f * cosf(TWO_PI * (float)n / 1024.f);
}

__global__ void build_wsuminv(const float* __restrict__ W, float* __restrict__ wsi) {
    int i = blockIdx.x * 256 + threadIdx.x;
    if (i >= YLEN) return;
    int tmax = i >> 8; if (tmax > TFRM - 1) tmax = TFRM - 1;
    int tmin = (i > NFFT - 1) ? ((i - 768) >> 8) : 0;
    float s = 0.f;
    for (int t = tmin; t <= tmax; ++t) { int n = i - (t << 8); s += W[n] * W[n]; }
    wsi[i] = 1.f / fmaxf(s, 1e-11f);
}

// ------------------------ per-batch elementwise stages ----------------------
__global__ void prep_mel(const float* __restrict__ lm, float* __restrict__ melT) {
    int e = blockIdx.x * 256 + threadIdx.x;             // < T*NMEL
    int b = blockIdx.y;
    int t = e / NMEL, m = e % NMEL;
    float x = lm[(long long)b * NMEL * TFRM + (long long)m * TFRM + t];
    melT[(long long)b * TFRM * NMEL + e] = expf(2.302585092994046f * 0.05f * x);
}

__global__ void init_angles(const float* __restrict__ are, const float* __restrict__ aim,
                            float* __restrict__ ang, float* __restrict__ tprev) {
    int e = blockIdx.x * 256 + threadIdx.x;             // < T*CPAD
    int b = blockIdx.y;
    int t = e / CPAD, c = e % CPAD;
    float v = 0.f;
    if (c < NFREQ)            v = are[(long long)b * NFREQ * TFRM + (long long)c * TFRM + t];
    else if (c < 2 * NFREQ)   v = aim[(long long)b * NFREQ * TFRM + (long long)(c - NFREQ) * TFRM + t];
    long long idx = (long long)b * TFRM * CPAD + e;
    ang[idx] = v;
    tprev[idx] = 0.f;
}

__global__ void mul_spec(const float* __restrict__ spec, const float* __restrict__ ang,
                         float* __restrict__ SA) {
    int e = blockIdx.x * 256 + threadIdx.x;             // < T*CPAD
    int b = blockIdx.y;
    int t = e / CPAD, c = e % CPAD;
    long long abase = (long long)b * TFRM * CPAD + e;
    float v = 0.f;
    if (c < 2 * NFREQ) {
        int f = (c < NFREQ) ? c : c - NFREQ;
        v = spec[(long long)b * TFRM * FPAD + (long long)t * FPAD + f] * ang[abase];
    }
    SA[abase] = v;
}

// overlap-add (window folded in) + normalization
__global__ void ola_norm(const float* __restrict__ Fr, const float* __restrict__ W,
                         const float* __restrict__ wsi, float* __restrict__ y) {
    int i = blockIdx.x * 256 + threadIdx.x;
    if (i >= YLEN) return;
    const float* fr = Fr + (long long)blockIdx.y * TFRM * NFFT;
    int tmax = i >> 8; if (tmax > TFRM - 1) tmax = TFRM - 1;
    int tmin = (i > NFFT - 1) ? ((i - 768) >> 8) : 0;
    float s = 0.f;
    for (int t = tmin; t <= tmax; ++t) { int n = i - (t << 8); s += fr[t * NFFT + n] * W[n]; }
    y[(long long)blockIdx.y * YLEN + i] = s * wsi[i];
}

// trim + reflect-pad + re-frame + window (fused) for the stft
__global__ void frame_win(const float* __restrict__ y, const float* __restrict__ W,
                          float* __restrict__ Fr) {
    int e = blockIdx.x * 256 + threadIdx.x;             // < T*NFFT
    int b = blockIdx.y;
    int t = e >> 10, n = e & 1023;
    int i = (t << 8) + n;
    int idx2 = (i < 512) ? (1024 - i)
             : (i < 512 + OUTL) ? i
             : (2 * OUTL + 1022 - i);
    Fr[(long long)b * TFRM * NFFT + e] = W[n] * y[(long long)b * YLEN + idx2];
}

// GriffinLim momentum phase update
__global__ void phase_update(const float* __restrict__ SA, float* __restrict__ tprev,
                             float* __restrict__ ang) {
    int e = blockIdx.x * 256 + threadIdx.x;             // < T*NFREQ
    int b = blockIdx.y;
    int t = e / NFREQ, f = e % NFREQ;
    long long base = (long long)b * TFRM * CPAD + (long long)t * CPAD + f;
    float re = SA[base], im = SA[base + NFREQ];
    float ar = re - MOMC * tprev[base];
    float ai = im - MOMC * tprev[base + NFREQ];
    float nrm = sqrtf(ar * ar + ai * ai) + 1e-16f;
    ang[base] = ar / nrm;
    ang[base + NFREQ] = ai / nrm;
    tprev[base] = re;
    tprev[base + NFREQ] = im;
}

__global__ void write_out(const float* __restrict__ y, float* __restrict__ out) {
    int e = blockIdx.x * 256 + threadIdx.x;
    if (e >= OUTL) return;
    int b = blockIdx.y;
    out[(long long)b * OUTL + e] = y[(long long)b * YLEN + 512 + e];
}

// ---------------------------------------------------------------------------
extern "C" void kernel_launch(void* const* d_in, const int* in_sizes, int n_in,
                              void* d_out, int out_size, void* d_ws, size_t ws_size,
                              hipStream_t stream) {
    (void)in_sizes; (void)n_in; (void)out_size; (void)ws_size;
    const float* lm  = (const float*)d_in[0];
    const float* are = (const float*)d_in[1];
    const float* aim = (const float*)d_in[2];
    float* out = (float*)d_out;

    char* ws = (char*)d_ws;
    size_t off = 0;
    auto take = [&](size_t nflt) -> float* {
        float* p = (float*)(ws + off);
        off += ((nflt * 4 + 255) & ~(size_t)255);
        return p;
    };
    float* fb    = take((size_t)NFREQ * NMEL);
    float* invG  = take((size_t)NMEL * NMEL);
    float* melB  = take((size_t)NMEL * FPAD);
    float* Cf    = take((size_t)NFFT * CPAD);
    float* Ci    = take((size_t)CPAD * NFFT);
    float* W     = take((size_t)NFFT);
    float* wsi   = take((size_t)YLEN);
    float* melT  = take((size_t)BATCH * TFRM * NMEL);
    float* spec  = take((size_t)BATCH * TFRM * FPAD);
    float* ang   = take((size_t)BATCH * TFRM * CPAD);
    float* tprev = take((size_t)BATCH * TFRM * CPAD);
    float* SA    = take((size_t)BATCH * TFRM * CPAD);
    float* Fr    = take((size_t)BATCH * TFRM * NFFT);
    float* yb    = take((size_t)BATCH * YLEN);

    // ---- setup (cheap, deterministic; rebuilt every call) ----
    build_fb     <<<(NFREQ * NMEL + 255) / 256, 256, 0, stream>>>(fb);
    invert_gram  <<<1, 256, 0, stream>>>(fb, invG);
    make_melB    <<<(NMEL * FPAD + 255) / 256, 256, 0, stream>>>(fb, invG, melB);
    build_cf     <<<(NFFT * CPAD) / 256, 256, 0, stream>>>(Cf);
    build_ci     <<<(CPAD * NFFT) / 256, 256, 0, stream>>>(Ci);
    build_window <<<4, 256, 0, stream>>>(W);
    build_wsuminv<<<YLEN / 256, 256, 0, stream>>>(W, wsi);

    // ---- mel -> linear magnitude spectrogram (WMMA GEMM, relu epilogue) ----
    prep_mel<<<dim3((TFRM * NMEL) / 256, BATCH), 256, 0, stream>>>(lm, melT);
    wmma_gemm_f32<<<dim3(FPAD / 64, TFRM / 64, BATCH), 128, 0, stream>>>(
        melT, melB, spec, TFRM, FPAD, NMEL, NMEL, FPAD, FPAD,
        (long long)TFRM * NMEL, (long long)TFRM * FPAD, 1);

    init_angles<<<dim3((TFRM * CPAD) / 256, BATCH), 256, 0, stream>>>(are, aim, ang, tprev);

    // ---- 32 Griffin-Lim iterations + final istft ----
    for (int it = 0; it <= 32; ++it) {
        mul_spec<<<dim3((TFRM * CPAD) / 256, BATCH), 256, 0, stream>>>(spec, ang, SA);
        // istft GEMM: frames = SA(T x CPAD) @ Ci(CPAD x NFFT)
        wmma_gemm_f32<<<dim3(NFFT / 64, TFRM / 64, BATCH), 128, 0, stream>>>(
            SA, Ci, Fr, TFRM, NFFT, CPAD, CPAD, NFFT, NFFT,
            (long long)TFRM * CPAD, (long long)TFRM * NFFT, 0);
        ola_norm<<<dim3(YLEN / 256, BATCH), 256, 0, stream>>>(Fr, W, wsi, yb);
        if (it == 32) break;
        frame_win<<<dim3((TFRM * NFFT) / 256, BATCH), 256, 0, stream>>>(yb, W, Fr);
        // stft GEMM: rebuilt = Fr(T x NFFT) @ Cf(NFFT x CPAD)
        wmma_gemm_f32<<<dim3(CPAD / 64, TFRM / 64, BATCH), 128, 0, stream>>>(
            Fr, Cf, SA, TFRM, CPAD, NFFT, NFFT, CPAD, CPAD,
            (long long)TFRM * NFFT, (long long)TFRM * CPAD, 0);
        phase_update<<<dim3((TFRM * NFREQ) / 256, BATCH), 256, 0, stream>>>(SA, tprev, ang);
    }

    write_out<<<dim3(OUTL / 256, BATCH), 256, 0, stream>>>(yb, out);
}